// diffPool_14139032339211
// MI455X (gfx1250) — compile-verified
//
#include <hip/hip_runtime.h>

// DiffPool-style pipeline for MI455X (gfx1250, wave32).
// HBM-bound: two streaming passes over the 268MB adjacency using
// v_wmma_f32_16x16x4_f32 for the dense SpMM, non-temporal loads for the adj
// stream, edge extraction fused into pass 1 so cluster adjacency is sparse.

#define N_NODES 8192
#define D 32
#define KCL 128
#define KSEG 4
#define KSEG_LEN (N_NODES / KSEG)   // 2048 columns of adj per K-segment
#define TILE_K 64
#define MAT ((size_t)N_NODES * D)   // 262144 elements

typedef __attribute__((ext_vector_type(2))) float v2f;
typedef __attribute__((ext_vector_type(8))) float v8f;

// ---------------------------------------------------------------------------
// Pass over adj: outPart[kseg] (+)= adj[:, kseg-range] @ B      (B is [N,32])
// Each wave owns 16 rows; block = 8 waves = 128 rows; blockIdx.y = K segment.
// EXTRACT=1 additionally appends off-diagonal nonzeros to a global edge list.
// ---------------------------------------------------------------------------
template <int EXTRACT>
__global__ __launch_bounds__(256) void k_spmm(const float* __restrict__ adj,
                                              const float* __restrict__ B,
                                              float* __restrict__ outPart,
                                              int2* __restrict__ edges,
                                              unsigned* __restrict__ ecnt,
                                              unsigned ecap) {
  __shared__ float tileA[8][16 * 65];   // 65-stride pad kills bank conflicts
  __shared__ float xs[TILE_K * 33];     // staged B slab, 33-stride pad

  const int tid  = threadIdx.x;
  const int lane = tid & 31;
  const int wave = tid >> 5;
  const int li   = lane & 15;           // N / M sub-index
  const int lh   = lane >> 4;           // half-wave select
  const int rowBase = blockIdx.x * 128 + wave * 16;
  const int kBeg = blockIdx.y * KSEG_LEN;
  const int kEnd = kBeg + KSEG_LEN;

  v8f c0 = {};  // output cols 0..15
  v8f c1 = {};  // output cols 16..31

  for (int kb = kBeg; kb < kEnd; kb += TILE_K) {
    // Stage B slab [kb..kb+64) x 32 into LDS (block cooperative, coalesced).
    for (int i = tid; i < TILE_K * D; i += 256) {
      int r = i >> 5, c = i & 31;
      xs[r * 33 + c] = B[(size_t)(kb + r) * D + c];
    }
    // Stage this wave's 16x64 adj tile (non-temporal: pure stream, 268MB).
    for (int r = 0; r < 16; ++r) {
      const int row = rowBase + r;
      const float* ap = adj + (size_t)row * N_NODES + kb + lane * 2;
      v2f v = __builtin_nontemporal_load((const v2f*)ap);
      tileA[wave][r * 65 + lane * 2]     = v.x;
      tileA[wave][r * 65 + lane * 2 + 1] = v.y;
      if (EXTRACT) {  // fuse edge extraction into the one mandatory adj pass
        const int col = kb + lane * 2;
        if (v.x != 0.0f && col != row) {
          unsigned e = atomicAdd(ecnt, 1u);
          if (e < ecap) edges[e] = make_int2(row, col);
        }
        if (v.y != 0.0f && col + 1 != row) {
          unsigned e = atomicAdd(ecnt, 1u);
          if (e < ecap) edges[e] = make_int2(row, col + 1);
        }
      }
      if (kb + TILE_K < kEnd) __builtin_prefetch(ap + TILE_K, 0, 0);
    }
    __syncthreads();

    const float* tA = tileA[wave];
#pragma unroll
    for (int k4 = 0; k4 < TILE_K / 4; ++k4) {
      const int kq = k4 * 4;
      v2f a, b0, b1;
      // A 16x4 fp32 layout: lanes0-15 K0/K1 rows M=li; lanes16-31 K2/K3.
      a.x = tA[li * 65 + kq + 2 * lh];
      a.y = tA[li * 65 + kq + 2 * lh + 1];
      // B 4x16 fp32 layout: VGPR0 lanes0-15=K0,lanes16-31=K1; VGPR1 K2/K3.
      b0.x = xs[(kq + lh) * 33 + li];
      b0.y = xs[(kq + 2 + lh) * 33 + li];
      b1.x = xs[(kq + lh) * 33 + 16 + li];
      b1.y = xs[(kq + 2 + lh) * 33 + 16 + li];
      c0 = __builtin_amdgcn_wmma_f32_16x16x4_f32(false, a, false, b0,
                                                 (short)0, c0, false, false);
      c1 = __builtin_amdgcn_wmma_f32_16x16x4_f32(false, a, false, b1,
                                                 (short)0, c1, false, false);
    }
    __syncthreads();
  }

  // C/D layout: VGPR v -> M = v (lanes0-15) / v+8 (lanes16-31), N = li.
  float* dst = outPart + (size_t)blockIdx.y * MAT;
#pragma unroll
  for (int vv = 0; vv < 8; ++vv) {
    const int row = rowBase + vv + 8 * lh;
    dst[(size_t)row * D + li]      = c0[vv];
    dst[(size_t)row * D + 16 + li] = c1[vv];
  }
}

// h = (sum_k aggPart) @ gc_w + gc_b   (fixed-order partial reduction)
__global__ __launch_bounds__(256) void k_linear(const float* __restrict__ aggPart,
                                                const float* __restrict__ W,
                                                const float* __restrict__ b,
                                                float* __restrict__ h) {
  __shared__ float Ws[D * D];
  __shared__ float bs[D];
  const int tid = threadIdx.x;
  for (int i = tid; i < D * D; i += 256) Ws[i] = W[i];
  if (tid < D) bs[tid] = b[tid];
  __syncthreads();
  const int node = blockIdx.x * 256 + tid;
  float r[D];
#pragma unroll
  for (int k = 0; k < D; ++k) {
    const size_t o = (size_t)node * D + k;
    r[k] = aggPart[o] + aggPart[MAT + o] + aggPart[2 * MAT + o] +
           aggPart[3 * MAT + o];
  }
#pragma unroll
  for (int o = 0; o < D; ++o) {
    float acc = bs[o];
#pragma unroll
    for (int k = 0; k < D; ++k) acc += r[k] * Ws[k * D + o];
    h[(size_t)node * D + o] = acc;
  }
}

// BatchNorm stats (single block, fixed-order tree reduce -> deterministic).
__global__ __launch_bounds__(256) void k_bn(const float* __restrict__ h,
                                            const float* __restrict__ gamma,
                                            const float* __restrict__ beta,
                                            float* __restrict__ coefA,
                                            float* __restrict__ coefB) {
  __shared__ float ssum[256], ssq[256];
  const int t = threadIdx.x;
  const int c = t & 31, chunk = t >> 5;  // 8 chunks of 1024 rows
  float s = 0.0f, q = 0.0f;
  for (int i = chunk * 1024; i < (chunk + 1) * 1024; ++i) {
    const float v = h[(size_t)i * D + c];
    s += v; q += v * v;
  }
  ssum[t] = s; ssq[t] = q;
  __syncthreads();
  if (t < D) {
    float S = 0.0f, Q = 0.0f;
    for (int k = 0; k < 8; ++k) { S += ssum[k * 32 + t]; Q += ssq[k * 32 + t]; }
    const float mu  = S * (1.0f / N_NODES);
    const float var = Q * (1.0f / N_NODES) - mu * mu;
    const float a = gamma[t] * rsqrtf(var + 1e-5f);
    coefA[t] = a;
    coefB[t] = beta[t] - mu * a;
  }
}

__global__ __launch_bounds__(256) void k_hn(const float* __restrict__ h,
                                            const float* __restrict__ coefA,
                                            const float* __restrict__ coefB,
                                            float* __restrict__ hn) {
  const int idx = blockIdx.x * 256 + threadIdx.x;
  const int c = idx & 31;
  hn[idx] = h[idx] * coefA[c] + coefB[c];
}

// argmax over softmax logits (softmax is monotone; first-max tie-break).
__global__ __launch_bounds__(256) void k_assign(const float* __restrict__ hnaggPart,
                                                const float* __restrict__ asw,
                                                const float* __restrict__ asb,
                                                int* __restrict__ sl) {
  __shared__ float Ws[D * KCL];
  __shared__ float bs[KCL];
  const int tid = threadIdx.x;
  for (int i = tid; i < D * KCL; i += 256) Ws[i] = asw[i];
  if (tid < KCL) bs[tid] = asb[tid];
  __syncthreads();
  const int node = blockIdx.x * 256 + tid;
  float r[D];
#pragma unroll
  for (int k = 0; k < D; ++k) {
    const size_t o = (size_t)node * D + k;
    r[k] = hnaggPart[o] + hnaggPart[MAT + o] + hnaggPart[2 * MAT + o] +
           hnaggPart[3 * MAT + o];
  }
  float best = -__builtin_huge_valf();
  int bi = 0;
  for (int c = 0; c < KCL; ++c) {
    float acc = bs[c];
#pragma unroll
    for (int k = 0; k < D; ++k) acc += r[k] * Ws[k * KCL + c];
    if (acc > best) { best = acc; bi = c; }
  }
  sl[node] = bi;
}

// Cluster adjacency from the sparse edge list (atomicOr: order-insensitive).
__global__ void k_cluster_adj(const int2* __restrict__ edges,
                              const unsigned* __restrict__ ecnt, unsigned ecap,
                              const int* __restrict__ sl,
                              int* __restrict__ Cmask) {
  unsigned ec = *ecnt;
  if (ec > ecap) ec = ecap;
  const unsigned stride = gridDim.x * blockDim.x;
  for (unsigned e = blockIdx.x * blockDim.x + threadIdx.x; e < ec; e += stride) {
    const int2 ed = edges[e];
    atomicOr(&Cmask[sl[ed.x] * KCL + sl[ed.y]], 1);
  }
}

// Per-cluster mean by gather (deterministic order; sl broadcast per wave).
__global__ __launch_bounds__(256) void k_pool(const float* __restrict__ hn,
                                              const int* __restrict__ sl,
                                              float* __restrict__ outPooled) {
  const int idx = blockIdx.x * 256 + threadIdx.x;  // 4096 = KCL*D
  const int s = idx >> 5, c = idx & 31;
  float acc = 0.0f;
  int cnt = 0;
  for (int i = 0; i < N_NODES; ++i) {
    if (sl[i] == s) { acc += hn[(size_t)i * D + c]; ++cnt; }
  }
  outPooled[idx] = acc / fmaxf((float)cnt, 1.0f);
}

__global__ __launch_bounds__(256) void k_final(const int* __restrict__ Cmask,
                                               float* __restrict__ outAdj) {
  const int idx = blockIdx.x * 256 + threadIdx.x;  // 16384
  const int c = idx >> 7, d = idx & 127;
  outAdj[idx] = (c == d) ? 1.0f : (Cmask[idx] ? 1.0f : 0.0f);
}

__global__ void zero_u32(unsigned* __restrict__ p, int n) {
  for (int i = blockIdx.x * blockDim.x + threadIdx.x; i < n;
       i += gridDim.x * blockDim.x)
    p[i] = 0u;
}

extern "C" void kernel_launch(void* const* d_in, const int* in_sizes, int n_in,
                              void* d_out, int out_size, void* d_ws,
                              size_t ws_size, hipStream_t stream) {
  (void)in_sizes; (void)n_in; (void)out_size;
  const float* x     = (const float*)d_in[0];
  const float* adj   = (const float*)d_in[1];
  const float* gcw   = (const float*)d_in[2];
  const float* gcb   = (const float*)d_in[3];
  const float* gamma = (const float*)d_in[4];
  const float* beta  = (const float*)d_in[5];
  const float* asw   = (const float*)d_in[6];
  const float* asb   = (const float*)d_in[7];
  float* out = (float*)d_out;

  // Workspace layout (fixed-order partials for determinism).
  char* w = (char*)d_ws;
  float* aggPart   = (float*)w;               // KSEG * MAT
  float* hnaggPart = aggPart + KSEG * MAT;    // KSEG * MAT
  float* h         = hnaggPart + KSEG * MAT;  // MAT
  float* hn        = h + MAT;                 // MAT
  float* coefA     = hn + MAT;                // 32
  float* coefB     = coefA + 32;              // 32
  int*   sl        = (int*)(coefB + 32);      // 8192
  int*   Cmask     = sl + N_NODES;            // 16384
  unsigned* ecnt   = (unsigned*)(Cmask + KCL * KCL);  // 1 (contig after Cmask)
  size_t used = (size_t)((char*)(ecnt + 1) - w);
  used = (used + 15) & ~(size_t)15;
  int2* edges = (int2*)(w + used);
  size_t ecap_sz = (ws_size > used) ? (ws_size - used) / sizeof(int2) : 0;
  unsigned ecap = (ecap_sz > 0x7FFFFFFFu) ? 0x7FFFFFFFu : (unsigned)ecap_sz;

  // Only accumulated-into buffers need zeroing each call.
  zero_u32<<<64, 256, 0, stream>>>((unsigned*)Cmask, KCL * KCL + 1);

  // Pass 1: agg = adj @ x (WMMA) + edge extraction.  268 MB HBM stream.
  k_spmm<1><<<dim3(N_NODES / 128, KSEG), 256, 0, stream>>>(
      adj, x, aggPart, edges, ecnt, ecap);
  // GraphConv linear + bias.
  k_linear<<<N_NODES / 256, 256, 0, stream>>>(aggPart, gcw, gcb, h);
  // BatchNorm (training-mode batch stats, biased variance).
  k_bn<<<1, 256, 0, stream>>>(h, gamma, beta, coefA, coefB);
  k_hn<<<MAT / 256, 256, 0, stream>>>(h, coefA, coefB, hn);
  // Pass 2: hn_agg = adj @ hn (WMMA).  Second 268 MB HBM stream.
  k_spmm<0><<<dim3(N_NODES / 128, KSEG), 256, 0, stream>>>(
      adj, hn, hnaggPart, nullptr, nullptr, 0u);
  // Cluster assignment, cluster adjacency, pooling, output.
  k_assign<<<N_NODES / 256, 256, 0, stream>>>(hnaggPart, asw, asb, sl);
  k_cluster_adj<<<256, 256, 0, stream>>>(edges, ecnt, ecap, sl, Cmask);
  k_pool<<<(KCL * D) / 256, 256, 0, stream>>>(hn, sl, out);
  k_final<<<(KCL * KCL) / 256, 256, 0, stream>>>(Cmask, out + KCL * D);
}